// AttentiveEncoder_73469710566059
// MI455X (gfx1250) — compile-verified
//
#include <hip/hip_runtime.h>
#include <cstdint>

// ---------------------------------------------------------------------------
// out[n,d] = w3[d] * relu(w2[d] * relu(w1[d] * relu(w0[d] * x[n,d])))
// Pure streaming elementwise op: fuse all 4 layers into one read+write pass.
// Data movement: CDNA5 Tensor Data Mover (tensor_load_to_lds), double-buffered
// in LDS, synchronized with s_wait_tensorcnt + workgroup barriers.
// Roofline: 200 MB read + 200 MB write @ 23.3 TB/s ≈ 17 us; compute is noise,
// so the only job is saturating HBM with one fused pass.
// ---------------------------------------------------------------------------

typedef unsigned int v4u __attribute__((ext_vector_type(4)));
typedef int          v8i __attribute__((ext_vector_type(8)));
typedef int          v4i __attribute__((ext_vector_type(4)));
typedef float        v4f __attribute__((ext_vector_type(4)));

#define DCOLS     1024      // feature dim (matches reference)
#define TILE_ROWS 8         // rows per TDM tile: 8*1024*4B = 32 KB per buffer

#if defined(__HIP_DEVICE_COMPILE__) && \
    __has_builtin(__builtin_amdgcn_tensor_load_to_lds) && \
    __has_builtin(__builtin_amdgcn_s_wait_tensorcnt)
#define USE_TDM 1
#else
#define USE_TDM 0
#endif

__device__ static inline v4f relu4(v4f a) {
    v4f r;
    r.x = fmaxf(a.x, 0.0f);
    r.y = fmaxf(a.y, 0.0f);
    r.z = fmaxf(a.z, 0.0f);
    r.w = fmaxf(a.w, 0.0f);
    return r;
}

#if USE_TDM
// Issue one TDM DMA: copy a (rows x 1024) f32 tile (row-major, stride 1024)
// from global `gsrc` into LDS byte-offset `lds_addr`.  Descriptor bitfields
// per CDNA5 ISA ch.8 (D# groups).  rows_remaining bounds OOB reads (-> zero).
__device__ static inline void tdm_load_tile(uint32_t lds_addr,
                                            const float* gsrc,
                                            uint32_t rows_remaining)
{
    const uint64_t ga = (uint64_t)(uintptr_t)gsrc;

    v4u g0;
    g0.x = 1u;                                   // count=1, user descriptor
    g0.y = lds_addr;                             // lds_addr [63:32]
    g0.z = (uint32_t)(ga & 0xFFFFFFFFu);         // global_addr[31:0]
    g0.w = (uint32_t)((ga >> 32) & 0x1FFFFFFu)   // global_addr[56:32]
         | (2u << 30);                           // type=2 ("image")

    v8i g1;
    g1[0] = (int)(2u << 16);                     // data_size=2 -> 4 bytes
    g1[1] = (int)((DCOLS & 0xFFFFu) << 16);      // tensor_dim0[15:0] (=1024)
    g1[2] = (int)(((DCOLS >> 16) & 0xFFFFu)      // tensor_dim0[31:16]
         | ((rows_remaining & 0xFFFFu) << 16));  // tensor_dim1[15:0]
    g1[3] = (int)(((rows_remaining >> 16) & 0xFFFFu)
         | ((uint32_t)DCOLS << 16));             // tile_dim0 = 1024
    g1[4] = (int)(TILE_ROWS & 0xFFFFu);          // tile_dim1 = 8, tile_dim2 = 0
    g1[5] = (int)DCOLS;                          // tensor_dim0_stride lo32
    g1[6] = 0;                                   // stride0 hi16, dim1_stride lo16
    g1[7] = 0;

    v4i g2 = {0, 0, 0, 0};                       // 2-D tensor: groups 2/3 zero
    v4i g3 = {0, 0, 0, 0};
    v8i g4 = {0, 0, 0, 0, 0, 0, 0, 0};           // extra group (clang-23 6-arg
                                                 // form); unused with count=1

    __builtin_amdgcn_tensor_load_to_lds(g0, g1, g2, g3, g4, /*cpol=*/0);
}
#endif

__global__ __launch_bounds__(256, 2)
void diag_relu_chain_kernel(const float* __restrict__ x,
                            const float* __restrict__ w,
                            float* __restrict__ out,
                            int n_tiles, int nrows)
{
    const int tid = threadIdx.x;            // 256 threads = 8 wave32
    const int col = tid << 2;               // each thread owns 4 fixed columns

    // Per-layer weights for this thread's columns: loaded once, L2-resident.
    const v4f w0 = *(const v4f*)(w + 0 * DCOLS + col);
    const v4f w1 = *(const v4f*)(w + 1 * DCOLS + col);
    const v4f w2 = *(const v4f*)(w + 2 * DCOLS + col);
    const v4f w3 = *(const v4f*)(w + 3 * DCOLS + col);

#if USE_TDM
    __shared__ float lds[2][TILE_ROWS * DCOLS];   // 2 x 32 KB double buffer
    const int wave = tid >> 5;
    const uint32_t lds0 = (uint32_t)(uintptr_t)(&lds[0][0]);
    const uint32_t lds1 = (uint32_t)(uintptr_t)(&lds[1][0]);

    int t = blockIdx.x;
    // Prologue: DMA the first tile.  TDM ignores EXEC, so gate by wave id.
    if (wave == 0 && t < n_tiles) {
        tdm_load_tile(lds0, x + (size_t)t * TILE_ROWS * DCOLS,
                      (uint32_t)(nrows - t * TILE_ROWS));
    }

    int buf = 0;
    for (; t < n_tiles; t += gridDim.x) {
        const int t_next = t + (int)gridDim.x;
        if (wave == 0) {
            if (t_next < n_tiles) {
                // Kick next tile into the other buffer, then wait for current.
                tdm_load_tile(buf ? lds0 : lds1,
                              x + (size_t)t_next * TILE_ROWS * DCOLS,
                              (uint32_t)(nrows - t_next * TILE_ROWS));
                __builtin_amdgcn_s_wait_tensorcnt(1); // current tile landed
            } else {
                __builtin_amdgcn_s_wait_tensorcnt(0); // last outstanding tile
            }
        }
        __syncthreads();   // publish DMA'd tile to all 8 waves

        const float* src = buf ? &lds[1][0] : &lds[0][0];
        float* dst = out + (size_t)t * TILE_ROWS * DCOLS;
        int rows_here = nrows - t * TILE_ROWS;
        if (rows_here > TILE_ROWS) rows_here = TILE_ROWS;

#pragma unroll
        for (int r = 0; r < TILE_ROWS; ++r) {
            if (r < rows_here) {
                v4f v = *(const v4f*)(src + r * DCOLS + col);   // ds_load_b128
                v4f y = relu4(v * w0);
                y = relu4(y * w1);
                y = relu4(y * w2);
                y = y * w3;                                      // no final relu
                *(v4f*)(dst + r * DCOLS + col) = y;              // b128 store
            }
        }
        __syncthreads();   // all waves done reading buf before it is refilled
        buf ^= 1;
    }
#else
    // Fallback (toolchain without TDM builtin): direct streamed B128 loads.
    for (int t = blockIdx.x; t < n_tiles; t += gridDim.x) {
        const size_t base = (size_t)t * TILE_ROWS * DCOLS;
        int rows_here = nrows - t * TILE_ROWS;
        if (rows_here > TILE_ROWS) rows_here = TILE_ROWS;
#pragma unroll
        for (int r = 0; r < TILE_ROWS; ++r) {
            if (r < rows_here) {
                v4f v = *(const v4f*)(x + base + r * DCOLS + col);
                v4f y = relu4(v * w0);
                y = relu4(y * w1);
                y = relu4(y * w2);
                y = y * w3;
                *(v4f*)(out + base + r * DCOLS + col) = y;
            }
        }
    }
#endif
}

extern "C" void kernel_launch(void* const* d_in, const int* in_sizes, int n_in,
                              void* d_out, int out_size, void* d_ws, size_t ws_size,
                              hipStream_t stream)
{
    const float* x   = (const float*)d_in[0];   // (N, 1024) f32
    const float* w   = (const float*)d_in[1];   // (4, 1024) f32
    float*       out = (float*)d_out;           // (N, 1024) f32

    const int nrows   = in_sizes[0] / DCOLS;                    // 50000
    const int n_tiles = (nrows + TILE_ROWS - 1) / TILE_ROWS;    // 6250
    const int grid    = n_tiles < 2048 ? n_tiles : 2048;

    diag_relu_chain_kernel<<<dim3(grid), dim3(256), 0, stream>>>(
        x, w, out, n_tiles, nrows);
}